// SSEGLAH_52518860095897
// MI455X (gfx1250) — compile-verified
//
#include <hip/hip_runtime.h>
#include <hip/hip_bf16.h>
#include <stdint.h>

// ---------------------------------------------------------------------------
// Shapes fixed by setup_inputs(): L=8192, H=6, D=256, N=4, S=4, K=2.
// ---------------------------------------------------------------------------
#define HD_C 1536   // H*D
#define N_C  4      // experts
#define K_C  2      // top-k
#define N4_C (HD_C / 4)  // 384 16-byte chunks per row

// Native clang vector type (works with __builtin_nontemporal_store, unlike
// HIP's struct-based float4).
typedef float v4f __attribute__((ext_vector_type(4)));

// ---------------------------------------------------------------------------
// CDNA5 async global<->LDS copy helpers (ASYNCcnt-tracked DMA path).
// gfx1250 builtins take pointers to 16-byte int vectors:
//   arg0 = global (AS1) v4i*, arg1 = LDS (AS3) v4i*, then imm offset, imm cpol.
// Typed pointers built via integer casts (low 32 bits of a flat shared
// address == LDS byte offset per the CDNA5 aperture mapping).
// ---------------------------------------------------------------------------
typedef __attribute__((__vector_size__(16))) int v4i_t;
typedef __attribute__((address_space(1))) v4i_t* as1_v4i;
typedef __attribute__((address_space(3))) v4i_t* as3_v4i;

#define TO_AS1(p) ((as1_v4i)(unsigned long long)(uintptr_t)(p))
#define TO_AS3(p) ((as3_v4i)(unsigned)(uintptr_t)(p))

__device__ __forceinline__ void async_load_b128(const void* gsrc, void* lds_dst) {
#if __has_builtin(__builtin_amdgcn_global_load_async_to_lds_b128)
  __builtin_amdgcn_global_load_async_to_lds_b128(TO_AS1(gsrc), TO_AS3(lds_dst), 0, 0);
#else
  unsigned loff = (unsigned)(uintptr_t)lds_dst;
  asm volatile("global_load_async_to_lds_b128 %0, %1, off"
               :: "v"(loff), "v"(gsrc) : "memory");
#endif
}

__device__ __forceinline__ void async_store_b128(void* gdst, const void* lds_src) {
#if __has_builtin(__builtin_amdgcn_global_store_async_from_lds_b128)
  __builtin_amdgcn_global_store_async_from_lds_b128(TO_AS1(gdst), TO_AS3(lds_src), 0, 0);
#else
  unsigned loff = (unsigned)(uintptr_t)lds_src;
  asm volatile("global_store_async_from_lds_b128 %0, %1, off"
               :: "v"(gdst), "v"(loff) : "memory");
#endif
}

__device__ __forceinline__ void wait_asynccnt0() {
#if __has_builtin(__builtin_amdgcn_s_wait_asynccnt)
  __builtin_amdgcn_s_wait_asynccnt(0);
#else
  asm volatile("s_wait_asynccnt 0" ::: "memory");
#endif
}

// ---------------------------------------------------------------------------
// Bandwidth kernel (defined FIRST so the disasm snippet shows it).
// One block per sorted slot (L*K blocks, 256 threads = 8 wave32).
//  Phase A: fire async DMA loads for v/gk (ASYNCcnt pipeline starts filling)
//  Phase B: q/k rows: b128 load -> scale by routing weight -> NT b128 store
//           (overlaps with the in-flight DMA)
//  Phase C: s_wait_asynccnt 0 -> async stores LDS -> global for v/gk
// Sources stay regular-temporal (fit in 192MB L2, K=2 reuse); destinations
// are streamed nontemporally so they don't evict the sources.
// ---------------------------------------------------------------------------
__global__ void gather_kernel(const float* __restrict__ q,
                              const float* __restrict__ k,
                              const float* __restrict__ v,
                              const float* __restrict__ gk,
                              const int* __restrict__ gsorted,
                              const float* __restrict__ vsorted,
                              float* __restrict__ qs, float* __restrict__ ks,
                              float* __restrict__ vs, float* __restrict__ gks,
                              float* __restrict__ out_gidx) {
  const int i = blockIdx.x;
  const int src = gsorted[i];
  const float w = vsorted[i];
  if (threadIdx.x == 0) out_gidx[i] = (float)src;

  const v4f* q4 = (const v4f*)(q + (size_t)src * HD_C);
  const v4f* k4 = (const v4f*)(k + (size_t)src * HD_C);
  const v4f* v4 = (const v4f*)(v + (size_t)src * HD_C);
  const v4f* g4 = (const v4f*)(gk + (size_t)src * HD_C);
  v4f* qs4 = (v4f*)(qs + (size_t)i * HD_C);
  v4f* ks4 = (v4f*)(ks + (size_t)i * HD_C);
  v4f* vs4 = (v4f*)(vs + (size_t)i * HD_C);
  v4f* gs4 = (v4f*)(gks + (size_t)i * HD_C);

  __shared__ v4f sv[N4_C];
  __shared__ v4f sg[N4_C];

  // Phase A: kick off the v/gk DMA (pure copy, never touches VGPRs).
  for (int j = threadIdx.x; j < N4_C; j += blockDim.x) {
    async_load_b128(&v4[j], &sv[j]);
    async_load_b128(&g4[j], &sg[j]);
  }

  // Phase B: scaled q/k path through VGPRs while DMA is in flight.
  for (int j = threadIdx.x; j < N4_C; j += blockDim.x) {
    v4f a = q4[j] * w;
    __builtin_nontemporal_store(a, &qs4[j]);
    v4f b = k4[j] * w;
    __builtin_nontemporal_store(b, &ks4[j]);
  }

  // Phase C: drain the DMA loads, then stream LDS -> global.
  wait_asynccnt0();  // per-wave counter; uniform point (loop bound is uniform)
  for (int j = threadIdx.x; j < N4_C; j += blockDim.x) {
    async_store_b128(&vs4[j], &sv[j]);
    async_store_b128(&gs4[j], &sg[j]);
  }
  // S_ENDPGM's implicit wait-idle covers the outstanding async stores.
}

// ---------------------------------------------------------------------------
// Kernel 0: zero the (sample,expert) pair counters.
// ---------------------------------------------------------------------------
__global__ void zero_counts_kernel(int* counts, int n) {
  int i = blockIdx.x * blockDim.x + threadIdx.x;
  if (i < n) counts[i] = 0;
}

// ---------------------------------------------------------------------------
// Kernel 1: per-token softmax over N experts, stable top-K, mask, pair counts.
// ---------------------------------------------------------------------------
__global__ void softmax_topk_kernel(const float* __restrict__ e,
                                    const long long* __restrict__ cu,
                                    int L, int S,
                                    float* __restrict__ e_soft_out,
                                    float* __restrict__ mask_out,
                                    int* __restrict__ topk_idx,
                                    float* __restrict__ topk_val,
                                    int* __restrict__ counts) {
  int t = blockIdx.x * blockDim.x + threadIdx.x;
  if (t >= L) return;

  float x[N_C];
  float m = -INFINITY;
  #pragma unroll
  for (int j = 0; j < N_C; ++j) { x[j] = e[(size_t)t * N_C + j]; m = fmaxf(m, x[j]); }
  float sum = 0.f;
  #pragma unroll
  for (int j = 0; j < N_C; ++j) { x[j] = __expf(x[j] - m); sum += x[j]; }
  float inv = 1.f / sum;
  #pragma unroll
  for (int j = 0; j < N_C; ++j) {
    x[j] *= inv;
    e_soft_out[(size_t)t * N_C + j] = x[j];
    mask_out[(size_t)t * N_C + j] = 0.f;
  }

  // sample id via cu_seqlens scan (S is tiny)
  int s = 0;
  for (int j = 0; j < S; ++j) if ((long long)t >= cu[j]) s = j;

  bool taken[N_C];
  #pragma unroll
  for (int j = 0; j < N_C; ++j) taken[j] = false;

  #pragma unroll
  for (int k = 0; k < K_C; ++k) {
    int best = -1; float bv = -INFINITY;
    #pragma unroll
    for (int j = 0; j < N_C; ++j)
      if (!taken[j] && x[j] > bv) { bv = x[j]; best = j; }  // ties -> lowest idx
    taken[best] = true;
    topk_idx[t * K_C + k] = best;
    topk_val[t * K_C + k] = bv;
    mask_out[(size_t)t * N_C + best] = 1.f;
    atomicAdd(&counts[s * N_C + best], 1);
  }
}

// ---------------------------------------------------------------------------
// Kernel 2: serial prefix sum over S*N pair counts -> pair offsets; emit the
// `offsets` and `state_sizes` outputs.
// ---------------------------------------------------------------------------
__global__ void offsets_kernel(const int* __restrict__ counts,
                               int* __restrict__ pair_off,
                               float* __restrict__ out_offsets,
                               float* __restrict__ out_state_sizes,
                               int SN) {
  if (threadIdx.x != 0 || blockIdx.x != 0) return;
  int acc = 0;
  pair_off[0] = 0;
  out_offsets[0] = 0.f;
  for (int i = 0; i < SN; ++i) {
    out_state_sizes[i] = (float)counts[i];
    acc += counts[i];
    pair_off[i + 1] = acc;
    out_offsets[i + 1] = (float)acc;
  }
}

// ---------------------------------------------------------------------------
// Kernel 3: one block per (sample, expert) pair; ordered block-scan over that
// sample's tokens reproduces the ascending (sample, expert, relpos) argsort.
// ---------------------------------------------------------------------------
__global__ void rank_kernel(const long long* __restrict__ cu,
                            const int* __restrict__ topk_idx,
                            const float* __restrict__ topk_val,
                            const int* __restrict__ pair_off,
                            int* __restrict__ gsorted,
                            float* __restrict__ vsorted) {
  const int pair = blockIdx.x;
  const int s = pair / N_C, ex = pair % N_C;
  const long long beg = cu[s], end = cu[s + 1];

  __shared__ int scan[256];
  __shared__ int sbase;
  if (threadIdx.x == 0) sbase = pair_off[pair];
  __syncthreads();

  for (long long chunk = beg; chunk < end; chunk += blockDim.x) {
    long long t = chunk + (long long)threadIdx.x;
    int flag = 0, kk = 0;
    if (t < end) {
      #pragma unroll
      for (int k = 0; k < K_C; ++k)
        if (topk_idx[t * K_C + k] == ex) { flag = 1; kk = k; }
    }
    // Hillis-Steele inclusive scan over the block
    scan[threadIdx.x] = flag;
    __syncthreads();
    for (int off = 1; off < (int)blockDim.x; off <<= 1) {
      int vv = scan[threadIdx.x];
      int add = (threadIdx.x >= (unsigned)off) ? scan[threadIdx.x - off] : 0;
      __syncthreads();
      scan[threadIdx.x] = vv + add;
      __syncthreads();
    }
    if (flag) {
      int pos = sbase + scan[threadIdx.x] - 1;
      gsorted[pos] = (int)t;
      vsorted[pos] = topk_val[t * K_C + kk];
    }
    __syncthreads();
    if (threadIdx.x == 0) sbase += scan[blockDim.x - 1];
    __syncthreads();
  }
}

// ---------------------------------------------------------------------------
// Host launcher
// ---------------------------------------------------------------------------
extern "C" void kernel_launch(void* const* d_in, const int* in_sizes, int n_in,
                              void* d_out, int out_size, void* d_ws, size_t ws_size,
                              hipStream_t stream) {
  const float* q  = (const float*)d_in[0];
  const float* k  = (const float*)d_in[1];
  const float* v  = (const float*)d_in[2];
  const float* gk = (const float*)d_in[3];
  const float* e  = (const float*)d_in[4];
  const long long* cu = (const long long*)d_in[5];  // np.int64 per setup

  const int L  = in_sizes[4] / N_C;     // 8192
  const int S  = in_sizes[5] - 1;       // 4
  const int SN = S * N_C;               // 16
  const int LK = L * K_C;               // 16384

  // Workspace carve-up.
  char* ws = (char*)d_ws;
  int*   counts    = (int*)ws;                      ws += 64;
  int*   pair_off  = (int*)ws;                      ws += 128;
  int*   topk_idx  = (int*)ws;                      ws += (size_t)LK * sizeof(int);
  float* topk_val  = (float*)ws;                    ws += (size_t)LK * sizeof(float);
  int*   gsorted   = (int*)ws;                      ws += (size_t)LK * sizeof(int);
  float* vsorted   = (float*)ws;

  // Output carve-up (reference tuple order, flattened).
  const size_t QS = (size_t)LK * HD_C;
  float* out     = (float*)d_out;
  float* o_qs    = out;
  float* o_ks    = out + QS;
  float* o_vs    = out + 2 * QS;
  float* o_gks   = out + 3 * QS;
  float* o_esoft = out + 4 * QS;
  float* o_mask  = o_esoft + (size_t)L * N_C;
  float* o_off   = o_mask + (size_t)L * N_C;
  float* o_ss    = o_off + (SN + 1);
  float* o_gidx  = o_ss + SN;

  zero_counts_kernel<<<1, 32, 0, stream>>>(counts, SN);

  softmax_topk_kernel<<<(L + 255) / 256, 256, 0, stream>>>(
      e, cu, L, S, o_esoft, o_mask, topk_idx, topk_val, counts);

  offsets_kernel<<<1, 32, 0, stream>>>(counts, pair_off, o_off, o_ss, SN);

  rank_kernel<<<SN, 256, 0, stream>>>(cu, topk_idx, topk_val, pair_off,
                                      gsorted, vsorted);

  gather_kernel<<<LK, 256, 0, stream>>>(q, k, v, gk, gsorted, vsorted,
                                        o_qs, o_ks, o_vs, o_gks, o_gidx);
}